// ProceduralMemory_9577777070706
// MI455X (gfx1250) — compile-verified
//
#include <hip/hip_runtime.h>
#include <math.h>

// Shapes fixed by the reference
#define BS 8
#define PP 1024
#define DD 512
#define RR 16
#define RHO_C 0.99f

typedef __attribute__((ext_vector_type(2))) float v2f;
typedef __attribute__((ext_vector_type(8))) float v8f;
typedef __attribute__((ext_vector_type(2))) int   v2i;

static __device__ __forceinline__ v8f wmma_f32(v2f a, v2f b, v8f c) {
  // V_WMMA_F32_16X16X4_F32 : D = A(16x4) x B(4x16) + C(16x16), fp32 throughout
  return __builtin_amdgcn_wmma_f32_16x16x4_f32(
      /*neg_a=*/false, a, /*neg_b=*/false, b,
      /*c_mod=*/(short)0, c, /*reuse_a=*/false, /*reuse_b=*/false);
}

// CDNA5 async global->LDS copy (ASYNCcnt path).
// Per the round-2 diagnostic, param 1 is a *generic* pointer to int2 (the
// source), so the signature follows the __builtin_amdgcn_load_to_lds pattern:
// (src generic ptr, dst LDS as(3) ptr, imm offset, imm cpol).
#if __has_builtin(__builtin_amdgcn_global_load_async_to_lds_b64) && \
    __has_builtin(__builtin_amdgcn_s_wait_asynccnt)
#define USE_ASYNC_LDS 1
static __device__ __forceinline__ void async_g2l_b64(void* lds, const void* g) {
  __builtin_amdgcn_global_load_async_to_lds_b64(
      (v2i*)g,
      (__attribute__((address_space(3))) v2i*)lds,
      0, 0);
}
#else
#define USE_ASYNC_LDS 0
#endif

// ---------------------------------------------------------------------------
// Kernel 1: C[b,p,d] = sum_k A[b,p,k]*W[d,k] + bias[d],  A = X or X*H (fused)
// 64x64 tile, BK=32. LDS tiles stored as k-pair float2 so every WMMA fragment
// is one ds_load_b64.
// ---------------------------------------------------------------------------
__global__ __launch_bounds__(256)
void gemm_xwT_kernel(const float* __restrict__ X, const float* __restrict__ H,
                     const float* __restrict__ W, const float* __restrict__ bias,
                     float* __restrict__ C) {
  __shared__ v2f As2[64][17];   // [p][k-pair]  (pad 17 -> conflict-free frags)
  __shared__ v2f Wl2[16][66];   // [k-pair][d]  (pad 66 -> conflict-free stores)

  const int b    = blockIdx.z;
  const int p0   = blockIdx.y * 64;
  const int d0   = blockIdx.x * 64;
  const int tx   = threadIdx.x;          // lane 0..31
  const int wv   = threadIdx.y;          // wave 0..7
  const int tid  = wv * 32 + tx;

  const int m0   = (wv >> 1) * 16;       // M subtile
  const int n0a  = ((wv & 1) * 2) * 16;  // N subtiles n0a, n0a+16
  const int n0b  = n0a + 16;
  const int lm   = tx & 15;              // lane -> M (A) / N (B,C)
  const int half = tx >> 4;              // half-wave -> K-pair select

  const float* Xb = X + (size_t)b * PP * DD;
  const float* Hb = H ? (H + (size_t)b * PP * DD) : nullptr;

  v8f acc0 = {}; v8f acc1 = {};

  for (int k0 = 0; k0 < DD; k0 += 32) {
    #pragma unroll
    for (int e = 0; e < 4; ++e) {
      int idx = tid + e * 256;           // 0..1023 (64 rows x 16 k-pairs)
      int rr = idx >> 4, kp = idx & 15;
      const float* wg = &W[(size_t)(d0 + rr) * DD + k0 + kp * 2];
#if USE_ASYNC_LDS
      async_g2l_b64(&Wl2[kp][rr], wg);   // pure copy -> async DMA to LDS
#else
      Wl2[kp][rr] = *(const v2f*)wg;
#endif
      v2f xv = *(const v2f*)&Xb[(size_t)(p0 + rr) * DD + k0 + kp * 2];
      if (Hb) xv *= *(const v2f*)&Hb[(size_t)(p0 + rr) * DD + k0 + kp * 2];
      As2[rr][kp] = xv;
    }
#if USE_ASYNC_LDS
    __builtin_amdgcn_s_wait_asynccnt(0);
#endif
    __syncthreads();
    #pragma unroll
    for (int kk = 0; kk < 32; kk += 4) {
      int kp = (kk >> 1) + half;
      v2f a  = As2[m0 + lm][kp];         // (m=lm, k=kk+2*half, kk+2*half+1)
      v2f b0 = Wl2[kp][n0a + lm];        // (k pair, n=lm)
      v2f b1 = Wl2[kp][n0b + lm];
      acc0 = wmma_f32(a, b0, acc0);
      acc1 = wmma_f32(a, b1, acc1);
    }
    __syncthreads();
  }

  float* Cb = C + (size_t)b * PP * DD;
  const int colA = d0 + n0a + lm;
  const int colB = d0 + n0b + lm;
  const float ba = bias[colA];
  const float bb = bias[colB];
  const int rbase = p0 + m0 + (half << 3);   // +8 rows for upper half-wave
  #pragma unroll
  for (int j = 0; j < 8; ++j) {
    Cb[(size_t)(rbase + j) * DD + colA] = acc0[j] + ba;
    Cb[(size_t)(rbase + j) * DD + colB] = acc1[j] + bb;
  }
}

// ---------------------------------------------------------------------------
// Kernel 2: per-(b,p) row: normalize x & kc, 32 dots vs pm_K, softmax route,
// gate, y_pm = x_q * sum_i a_i (x_q . K_i) V_i, store g*route, normalize kc.
// float2 per thread (256 threads cover D=512).
// ---------------------------------------------------------------------------
__global__ __launch_bounds__(256)
void row_kernel(const float* __restrict__ x_all, const float* __restrict__ surprise,
                const float* __restrict__ pm_K, const float* __restrict__ pm_V,
                const float* __restrict__ pm_a,
                float* __restrict__ kc,     // in: raw k_cand, out: normalized
                float* __restrict__ gr,     // [BS,PP,RR] gate*route
                float* __restrict__ y) {    // [BS,PP,DD]
  const int p = blockIdx.x, b = blockIdx.y;
  const int tid = threadIdx.x;
  const int lane = tid & 31, wv = tid >> 5;
  const int d2 = tid * 2;

  const float* xr  = x_all + ((size_t)b * PP + p) * DD;
  float*       kcr = kc    + ((size_t)b * PP + p) * DD;
  const float* Kb  = pm_K  + (size_t)b * RR * DD;
  const float* Vb  = pm_V  + (size_t)b * RR * DD;

  const v2f x2 = *(const v2f*)&xr[d2];
  const v2f k2 = *(const v2f*)&kcr[d2];

  // partials: [0]=|x|^2 [1]=|kc|^2 [2..17]=x.K_i [18..33]=kc.K_i
  float part[34];
  part[0] = x2.x * x2.x + x2.y * x2.y;
  part[1] = k2.x * k2.x + k2.y * k2.y;
  #pragma unroll
  for (int i = 0; i < 16; ++i) {
    v2f K2 = *(const v2f*)&Kb[i * DD + d2];
    part[2 + i]  = x2.x * K2.x + x2.y * K2.y;
    part[18 + i] = k2.x * K2.x + k2.y * K2.y;
  }

  __shared__ float red[34][8];
  __shared__ float tot[34];
  #pragma unroll
  for (int q = 0; q < 34; ++q) {
    float v = part[q];
    #pragma unroll
    for (int off = 16; off > 0; off >>= 1) v += __shfl_xor(v, off, 32);
    if (lane == 0) red[q][wv] = v;
  }
  __syncthreads();
  if (tid < 34) {
    float s = 0.f;
    #pragma unroll
    for (int w = 0; w < 8; ++w) s += red[tid][w];
    tot[tid] = s;
  }
  __syncthreads();

  const float xn = fmaxf(sqrtf(tot[0]), 1e-6f);
  const float kn = fmaxf(sqrtf(tot[1]), 1e-6f);

  // softmax over route logits (tau = 1)
  float mx = -3.4e38f;
  #pragma unroll
  for (int i = 0; i < 16; ++i) mx = fmaxf(mx, tot[18 + i] / kn);
  float es = 0.f;
  #pragma unroll
  for (int i = 0; i < 16; ++i) es += expf(tot[18 + i] / kn - mx);

  const float g = fminf(fmaxf(surprise[(size_t)b * PP + p], 0.f), 1.f);
  if (tid < 16)
    gr[((size_t)b * PP + p) * RR + tid] = g * expf(tot[18 + tid] / kn - mx) / es;

  float coef[16];
  #pragma unroll
  for (int i = 0; i < 16; ++i) coef[i] = pm_a[b * RR + i] * (tot[2 + i] / xn);

  v2f acc = {};
  #pragma unroll
  for (int i = 0; i < 16; ++i) {
    v2f V2 = *(const v2f*)&Vb[i * DD + d2];
    acc.x += coef[i] * V2.x;
    acc.y += coef[i] * V2.y;
  }
  float* yr = y + ((size_t)b * PP + p) * DD;
  v2f yo; yo.x = (x2.x / xn) * acc.x; yo.y = (x2.y / xn) * acc.y;
  *(v2f*)&yr[d2] = yo;
  v2f ko; ko.x = k2.x / kn; ko.y = k2.y / kn;
  *(v2f*)&kcr[d2] = ko;                 // normalized k_cand for elig GEMM
}

// ---------------------------------------------------------------------------
// Kernel 3: suffix decay products  w_p = prod_{q>p} (rho * carry_q),
//           totals[b] = prod over all q (scales elig0)
// ---------------------------------------------------------------------------
__global__ void scan_kernel(const unsigned char* __restrict__ reset,
                            float* __restrict__ wsuf, float* __restrict__ totals) {
  int b = threadIdx.x;
  if (b >= BS) return;
  float acc = 1.f;
  for (int p = PP - 1; p >= 0; --p) {
    wsuf[b * PP + p] = acc;
    float c = reset[b * PP + p] ? 0.f : 1.f;
    acc *= RHO_C * c;
  }
  totals[b] = acc;
}

// ---------------------------------------------------------------------------
// Kernel 3b: awT[b,i,p] = wsuf[b,p] * gr[b,p,i]   (transposed for b64 A-frags)
// ---------------------------------------------------------------------------
__global__ __launch_bounds__(256)
void fold_kernel(const float* __restrict__ gr, const float* __restrict__ wsuf,
                 float* __restrict__ awT) {
  int idx = blockIdx.x * 256 + threadIdx.x;   // over BS*PP*RR = 2^17
  int i = idx & (RR - 1);
  int p = (idx >> 4) & (PP - 1);
  int b = idx >> 14;
  awT[((size_t)b * RR + i) * PP + p] = wsuf[b * PP + p] * gr[idx];
}

// ---------------------------------------------------------------------------
// Kernel 4: elig[b,i,d] = totals[b]*elig0[b,i,d] + sum_p awT[b,i,p]*cand[b,p,d]
// WMMA GEMM M=16(r) x N=2x16(d) x K=1024(p); A-fragment reused across 2 tiles.
// ---------------------------------------------------------------------------
__global__ __launch_bounds__(128)
void elig_kernel(const float* __restrict__ awT, const float* __restrict__ totals,
                 const float* __restrict__ cand, const float* __restrict__ elig0,
                 float* __restrict__ out) {
  const int b  = blockIdx.y;
  const int d0 = blockIdx.x * 128 + threadIdx.y * 32;
  const int tx = threadIdx.x;
  const int lm = tx & 15;
  const int kh = (tx >> 4) * 2;

  const float* ab = awT + ((size_t)b * RR + lm) * PP;   // row i = lm
  const float* cb = cand + (size_t)b * PP * DD;

  v8f acc0 = {}, acc1 = {};
  #pragma unroll 4
  for (int p = 0; p < PP; p += 4) {
    const int pa = p + kh;
    v2f a = *(const v2f*)&ab[pa];                       // (i=lm, p=pa, pa+1)
    v2f b0; b0.x = cb[(size_t)pa * DD + d0 + lm];
            b0.y = cb[(size_t)(pa + 1) * DD + d0 + lm];
    v2f b1; b1.x = cb[(size_t)pa * DD + d0 + 16 + lm];
            b1.y = cb[(size_t)(pa + 1) * DD + d0 + 16 + lm];
    acc0 = wmma_f32(a, b0, acc0);
    acc1 = wmma_f32(a, b1, acc1);
  }

  const float t = totals[b];
  const float* e0 = elig0 + (size_t)b * RR * DD;
  float*       ob = out   + (size_t)b * RR * DD;
  const int rbase = (tx >> 4) << 3;
  #pragma unroll
  for (int j = 0; j < 8; ++j) {
    int row = rbase + j;
    int cA = d0 + lm, cB = d0 + 16 + lm;
    ob[(size_t)row * DD + cA] = t * e0[(size_t)row * DD + cA] + acc0[j];
    ob[(size_t)row * DD + cB] = t * e0[(size_t)row * DD + cB] + acc1[j];
  }
}

// ---------------------------------------------------------------------------
extern "C" void kernel_launch(void* const* d_in, const int* in_sizes, int n_in,
                              void* d_out, int out_size, void* d_ws, size_t ws_size,
                              hipStream_t stream) {
  const float* x_all    = (const float*)d_in[0];
  const float* h_all    = (const float*)d_in[1];
  const float* surprise = (const float*)d_in[2];
  const float* pm_K     = (const float*)d_in[3];
  const float* pm_V     = (const float*)d_in[4];
  const float* pm_a     = (const float*)d_in[5];
  const float* elig_K0  = (const float*)d_in[6];
  const float* elig_V0  = (const float*)d_in[7];
  const float* Wk_w     = (const float*)d_in[8];
  const float* Wk_b     = (const float*)d_in[9];
  const float* Wv_w     = (const float*)d_in[10];
  const float* Wv_b     = (const float*)d_in[11];
  const unsigned char* reset_mask = (const unsigned char*)d_in[12];

  // outputs concatenated: y_pm [BS,PP,DD], elig_K [BS,RR,DD], elig_V [BS,RR,DD]
  float* y_pm   = (float*)d_out;
  float* elig_K = y_pm + (size_t)BS * PP * DD;
  float* elig_V = elig_K + (size_t)BS * RR * DD;

  // workspace: kc + vc + gr + wsuf + totals + awT  (~35 MB)
  float* ws     = (float*)d_ws;
  float* kc     = ws;
  float* vc     = kc + (size_t)BS * PP * DD;
  float* gr     = vc + (size_t)BS * PP * DD;
  float* wsuf   = gr + (size_t)BS * PP * RR;
  float* totals = wsuf + (size_t)BS * PP;
  float* awT    = totals + 64;   // keep 256B alignment

  dim3 gg(DD / 64, PP / 64, BS), gb(32, 8);
  gemm_xwT_kernel<<<gg, gb, 0, stream>>>(x_all, nullptr, Wk_w, Wk_b, kc);
  gemm_xwT_kernel<<<gg, gb, 0, stream>>>(x_all, h_all,  Wv_w, Wv_b, vc);

  row_kernel<<<dim3(PP, BS), 256, 0, stream>>>(x_all, surprise, pm_K, pm_V, pm_a,
                                               kc, gr, y_pm);

  scan_kernel<<<1, 32, 0, stream>>>(reset_mask, wsuf, totals);
  fold_kernel<<<(BS * PP * RR) / 256, 256, 0, stream>>>(gr, wsuf, awT);

  dim3 eg(DD / 128, BS), eb(32, 4);
  elig_kernel<<<eg, eb, 0, stream>>>(awT, totals, kc, elig_K0, elig_K);
  elig_kernel<<<eg, eb, 0, stream>>>(awT, totals, vc, elig_V0, elig_V);
}